// DeformableSpatialEncoder_89618787598351
// MI455X (gfx1250) — compile-verified
//
#include <hip/hip_runtime.h>

typedef __attribute__((ext_vector_type(2))) float v2f;
typedef __attribute__((ext_vector_type(8))) float v8f;

#define IMG   224
#define PATCH 16
#define HF    14
#define WF    14
#define LQ    196
#define DM    768
#define NH    4
#define NP    4
#define DH    192
#define NIMG  64
#define MROWS (NIMG * LQ)   // 12544

// ---------------------------------------------------------------------------
// 1) im2col: x (N,3,224,224) -> A[MROWS][768], col = c*256 + p*16 + q
// ---------------------------------------------------------------------------
__global__ __launch_bounds__(256) void im2col_kernel(const float* __restrict__ x,
                                                     float* __restrict__ out) {
  long idx = (long)blockIdx.x * 256 + threadIdx.x;
  const long total = (long)MROWS * DM;
  if (idx >= total) return;
  int col = (int)(idx % DM);
  int row = (int)(idx / DM);
  int n = row / LQ, t = row % LQ;
  int h = t / WF, w = t % WF;
  int c = col >> 8, pq = col & 255, p = pq >> 4, q = pq & 15;
  out[idx] = x[(((long)(n * 3 + c) * IMG) + h * PATCH + p) * IMG + w * PATCH + q];
}

// ---------------------------------------------------------------------------
// 2) FP32 WMMA GEMM with async global->LDS double buffering.
//    C[M][N] = A[M][K] * W[N][K]^T + bias[N]
//    Block = 128 threads (4 waves). Per block: (64*AT) rows x 64 cols.
//    Each wave: AT A-fragments (16 rows each) x 4 N-tiles -> AT*4 accum tiles.
//    LDS layout: row-major [row][k] with stride 20 (pad) so every fragment
//    (K-pair) is one aligned ds_load_b64; halves of the wave hit disjoint
//    bank sets (residues 0 mod 4 vs 2 mod 4).
//    Requires: M % (64*AT) == 0, N % 64 == 0, K % 16 == 0, 16B-aligned ptrs.
// ---------------------------------------------------------------------------
template <int AT>
__global__ __launch_bounds__(128) void gemm_wmma_f32_async(
    const float* __restrict__ A, const float* __restrict__ W,
    const float* __restrict__ bias, float* __restrict__ C,
    int M, int K, int N) {
  __shared__ float As[2][64 * AT * 20];  // [buf][row*20 + k]
  __shared__ float Bs[2][64 * 20];       // [buf][n*20 + k]  (B stored n-major)

  const int tid  = threadIdx.x;
  const int wave = tid >> 5;
  const int lane = tid & 31;
  const int lr   = lane & 15;   // M (A) / N (B,C) index within fragment
  const int lh   = lane >> 4;   // half-wave selector
  const int m0   = blockIdx.y * (64 * AT);
  const int n0   = blockIdx.x * 64;

  // --- async tile fetch: one global_load_async_to_lds_b128 moves 4 floats/lane
  auto issue = [&](int buf, int k0) {
    // A tile: (64*AT) rows x 16 k -> (64*AT)*4 quad-slots, 2*AT per thread
#pragma unroll
    for (int i = 0; i < 2 * AT; ++i) {
      int slot = i * 128 + tid;
      int row = slot >> 2, kq = slot & 3;
      unsigned ga  = (unsigned)(((long)(m0 + row) * K + k0 + kq * 4) * 4);
      unsigned lds = (unsigned)(size_t)&As[buf][row * 20 + kq * 4];
      asm volatile("global_load_async_to_lds_b128 %0, %1, %2"
                   :: "v"(lds), "v"(ga), "s"(A) : "memory");
    }
    // B tile: 64 n-rows x 16 k -> 256 quad-slots, 2 per thread
#pragma unroll
    for (int i = 0; i < 2; ++i) {
      int slot = i * 128 + tid;
      int row = slot >> 2, kq = slot & 3;
      unsigned ga  = (unsigned)(((long)(n0 + row) * K + k0 + kq * 4) * 4);
      unsigned lds = (unsigned)(size_t)&Bs[buf][row * 20 + kq * 4];
      asm volatile("global_load_async_to_lds_b128 %0, %1, %2"
                   :: "v"(lds), "v"(ga), "s"(W) : "memory");
    }
  };

  v8f acc[AT][4];
#pragma unroll
  for (int a = 0; a < AT; ++a)
#pragma unroll
    for (int t = 0; t < 4; ++t)
      acc[a][t] = (v8f){0.f, 0.f, 0.f, 0.f, 0.f, 0.f, 0.f, 0.f};

  issue(0, 0);
  asm volatile("s_wait_asynccnt 0x0" ::: "memory");
  __syncthreads();

  const int nslab = K >> 4;
  for (int s = 0; s < nslab; ++s) {
    const int buf = s & 1;
    if (s + 1 < nslab) issue(buf ^ 1, (s + 1) * 16);

#pragma unroll
    for (int kk = 0; kk < 16; kk += 4) {
      const int ka = kk + 2 * lh;  // K index of fragment reg 0 (reg 1 = ka+1)
      v2f aF[AT];
#pragma unroll
      for (int a = 0; a < AT; ++a)
        aF[a] = *(const v2f*)&As[buf][(wave * (16 * AT) + a * 16 + lr) * 20 + ka];
#pragma unroll
      for (int t = 0; t < 4; ++t) {
        v2f bF = *(const v2f*)&Bs[buf][(t * 16 + lr) * 20 + ka];
#pragma unroll
        for (int a = 0; a < AT; ++a)
          acc[a][t] = __builtin_amdgcn_wmma_f32_16x16x4_f32(
              /*neg_a=*/false, aF[a], /*neg_b=*/false, bF,
              /*c_mod=*/(short)0, acc[a][t], /*reuse_a=*/false, /*reuse_b=*/false);
      }
    }

    asm volatile("s_wait_asynccnt 0x0" ::: "memory");
    __syncthreads();
  }

  // C/D layout: reg r, lane L -> M = r + 8*(L/16), N = L%16
#pragma unroll
  for (int a = 0; a < AT; ++a)
#pragma unroll
    for (int t = 0; t < 4; ++t)
#pragma unroll
      for (int r = 0; r < 8; ++r) {
        int m = m0 + wave * (16 * AT) + a * 16 + r + 8 * lh;
        int n = n0 + t * 16 + lr;
        C[(long)m * N + n] = acc[a][t][r] + bias[n];
      }
}

// ---------------------------------------------------------------------------
// 3) Offsets (32/row) + attention logits (16/row) + per-head softmax
// ---------------------------------------------------------------------------
__global__ __launch_bounds__(128) void offaw_kernel(const float* __restrict__ feat,
                                                    const float* __restrict__ off_w,
                                                    const float* __restrict__ off_b,
                                                    const float* __restrict__ aw_w,
                                                    const float* __restrict__ aw_b,
                                                    float* __restrict__ off,
                                                    float* __restrict__ aw) {
  __shared__ float fr[DM];
  __shared__ float logits[NH * NP];
  const int row = blockIdx.x, tid = threadIdx.x;

  for (int i = tid; i < DM; i += 128) fr[i] = feat[(long)row * DM + i];
  __syncthreads();

  if (tid < 32) {
    const float* wrow = off_w + (long)tid * DM;
    float s = off_b[tid];
    for (int k = 0; k < DM; ++k) s += fr[k] * wrow[k];
    off[(long)row * 32 + tid] = s;
  } else if (tid < 48) {
    int j = tid - 32;
    const float* wrow = aw_w + (long)j * DM;
    float s = aw_b[j];
    for (int k = 0; k < DM; ++k) s += fr[k] * wrow[k];
    logits[j] = s;
  }
  __syncthreads();

  if (tid < NH) {
    int h = tid;
    float m = logits[h * NP];
    for (int p = 1; p < NP; ++p) m = fmaxf(m, logits[h * NP + p]);
    float e[NP], sum = 0.f;
    for (int p = 0; p < NP; ++p) { e[p] = __expf(logits[h * NP + p] - m); sum += e[p]; }
    float inv = 1.f / sum;
    for (int p = 0; p < NP; ++p) aw[(long)row * (NH * NP) + h * NP + p] = e[p] * inv;
  }
}

// ---------------------------------------------------------------------------
// 4) Deformable-attention core: bilinear gather + weighted combine per token
// ---------------------------------------------------------------------------
__global__ __launch_bounds__(256) void deform_kernel(const float* __restrict__ value,
                                                     const float* __restrict__ off,
                                                     const float* __restrict__ aw,
                                                     float* __restrict__ attn_out) {
  __shared__ float sw[NH * NP][4];   // combined weight (aw * bilinear * valid)
  __shared__ int   sidx[NH * NP][4]; // clamped token index per corner
  const int row = blockIdx.x, tid = threadIdx.x;
  const int n = row / LQ, t = row % LQ;

  if (tid < NH * NP) {
    float refx = (float)(t % WF) / (float)(WF - 1);
    float refy = (float)(t / WF) / (float)(HF - 1);
    float ox = off[(long)row * 32 + tid * 2 + 0];
    float oy = off[(long)row * 32 + tid * 2 + 1];
    // loc = ref + off/(W,H);  pixel = loc*S - 0.5 (align_corners=False)
    float px = (refx + ox / (float)WF) * (float)WF - 0.5f;
    float py = (refy + oy / (float)HF) * (float)HF - 0.5f;
    float fx = floorf(px), fy = floorf(py);
    int x0 = (int)fx, y0 = (int)fy;
    float wx1 = px - fx, wy1 = py - fy;
    float wx0 = 1.f - wx1, wy0 = 1.f - wy1;
    float a = aw[(long)row * (NH * NP) + tid];
    int   xs[2]  = {x0, x0 + 1}, ys[2] = {y0, y0 + 1};
    float wxs[2] = {wx0, wx1},  wys[2] = {wy0, wy1};
#pragma unroll
    for (int cy = 0; cy < 2; ++cy) {
#pragma unroll
      for (int cx = 0; cx < 2; ++cx) {
        int  cI = cy * 2 + cx;
        bool valid = xs[cx] >= 0 && xs[cx] < WF && ys[cy] >= 0 && ys[cy] < HF;
        int  xi = min(max(xs[cx], 0), WF - 1);
        int  yi = min(max(ys[cy], 0), HF - 1);
        sw[tid][cI]   = valid ? a * wxs[cx] * wys[cy] : 0.f;
        sidx[tid][cI] = yi * WF + xi;
      }
    }
  }
  __syncthreads();

  const long vbase = (long)n * LQ;
  for (int d = tid; d < DM; d += 256) {
    int h = d / DH, c = d - h * DH;
    float s = 0.f;
#pragma unroll
    for (int p = 0; p < NP; ++p) {
      int e = h * NP + p;
#pragma unroll
      for (int cI = 0; cI < 4; ++cI)
        s += sw[e][cI] * value[(vbase + sidx[e][cI]) * DM + h * DH + c];
    }
    attn_out[(long)row * DM + d] = s;
  }
}

// ---------------------------------------------------------------------------
// 5) Mean pool over tokens (fused before output projection; linearity)
// ---------------------------------------------------------------------------
__global__ __launch_bounds__(256) void pool_kernel(const float* __restrict__ attn,
                                                   float* __restrict__ pooled) {
  const int n = blockIdx.x, tid = threadIdx.x;
  for (int d = tid; d < DM; d += 256) {
    float s = 0.f;
    for (int q = 0; q < LQ; ++q) s += attn[((long)n * LQ + q) * DM + d];
    pooled[(long)n * DM + d] = s * (1.f / (float)LQ);
  }
}

// ---------------------------------------------------------------------------
extern "C" void kernel_launch(void* const* d_in, const int* in_sizes, int n_in,
                              void* d_out, int out_size, void* d_ws, size_t ws_size,
                              hipStream_t stream) {
  const float* x       = (const float*)d_in[0];
  const float* embed_w = (const float*)d_in[1];
  const float* embed_b = (const float*)d_in[2];
  const float* value_w = (const float*)d_in[3];
  const float* value_b = (const float*)d_in[4];
  const float* off_w   = (const float*)d_in[5];
  const float* off_b   = (const float*)d_in[6];
  const float* aw_w    = (const float*)d_in[7];
  const float* aw_b    = (const float*)d_in[8];
  const float* outp_w  = (const float*)d_in[9];
  const float* outp_b  = (const float*)d_in[10];
  const float* proj_w  = (const float*)d_in[11];
  const float* proj_b  = (const float*)d_in[12];

  float* ws = (float*)d_ws;
  const long PF = (long)MROWS * DM;            // 9,633,792 floats per big buffer
  float* im2col  = ws;                         // A matrix (reused later as attn)
  float* feat    = ws + PF;
  float* value   = ws + 2 * PF;
  float* off     = ws + 3 * PF;
  float* aw      = off + (long)MROWS * 32;
  float* pooledA = aw + (long)MROWS * 16;      // 64 x 768
  float* pooled2 = pooledA + (long)NIMG * DM;  // 64 x 768
  float* attn    = im2col;                     // safe reuse: im2col dead after G1

  const long total = PF;
  im2col_kernel<<<(unsigned)((total + 255) / 256), 256, 0, stream>>>(x, im2col);

  dim3 gBig(DM / 64, MROWS / 128);  // AT=2 -> 128 rows per block
  gemm_wmma_f32_async<2><<<gBig, 128, 0, stream>>>(im2col, embed_w, embed_b, feat,
                                                   MROWS, DM, DM);
  gemm_wmma_f32_async<2><<<gBig, 128, 0, stream>>>(feat, value_w, value_b, value,
                                                   MROWS, DM, DM);

  offaw_kernel<<<MROWS, 128, 0, stream>>>(feat, off_w, off_b, aw_w, aw_b, off, aw);
  deform_kernel<<<MROWS, 256, 0, stream>>>(value, off, aw, attn);
  pool_kernel<<<NIMG, 256, 0, stream>>>(attn, pooledA);

  dim3 gSmall(DM / 64, NIMG / 64);  // AT=1 -> 64 rows per block
  gemm_wmma_f32_async<1><<<gSmall, 128, 0, stream>>>(pooledA, outp_w, outp_b, pooled2,
                                                     NIMG, DM, DM);
  gemm_wmma_f32_async<1><<<gSmall, 128, 0, stream>>>(pooled2, proj_w, proj_b,
                                                     (float*)d_out, NIMG, DM, DM);
}